// PredictiveSpanFinder_22282290331891
// MI455X (gfx1250) — compile-verified
//
#include <hip/hip_runtime.h>
#include <hip/hip_bf16.h>
#include <math.h>

// Problem constants (reference: B=2, L=512, D=1024, H=256)
#define BD 2
#define LL 512
#define DD 1024
#define HH 256
#define NROWS (BD * (LL + 1))   // 1026 rows of x = [null; seq]
#define NEGV (-1000.0f)
#define JSPLIT 8                // waves cooperating on one score row (j strided)

typedef __attribute__((ext_vector_type(2))) float v2f;
typedef __attribute__((ext_vector_type(8))) float v8f;

__device__ __forceinline__ v8f wmma_f32_k4(v2f a, v2f b, v8f c) {
  // D = A(16x4) x B(4x16) + C(16x16), fp32 — CDNA5 V_WMMA_F32_16X16X4_F32
  return __builtin_amdgcn_wmma_f32_16x16x4_f32(false, a, false, b, (short)0, c,
                                               false, false);
}

__device__ __forceinline__ float gelu_exact(float x) {
  return 0.5f * x * (1.0f + erff(x * 0.70710678118654752440f));
}

// ---------------------------------------------------------------------------
// K1: S = x @ Ws + bs ; E = x @ We + be   (x = [null_embed ; seq], row-major)
// One wave per 16x16 output tile; A tile shared between the two GEMMs.
// ---------------------------------------------------------------------------
__global__ __launch_bounds__(32)
void psf_proj1(const float* __restrict__ seq, const float* __restrict__ nullemb,
               const float* __restrict__ Ws, const float* __restrict__ bs,
               const float* __restrict__ We, const float* __restrict__ be,
               float* __restrict__ S, float* __restrict__ E) {
  const int lane = threadIdx.x & 31;
  const int half = lane >> 4;    // 0: K=0,1 ; 1: K=2,3
  const int l15  = lane & 15;
  const int m0 = blockIdx.x * 16;
  const int n0 = blockIdx.y * 16;

  // Resolve this lane's A row pointer once (implicit concat of null row)
  int g = m0 + l15;
  if (g >= NROWS) g = 0;         // padded rows compute garbage, never stored
  const int bb = g / (LL + 1);
  const int tt = g % (LL + 1);
  const float* xrow = (tt == 0) ? nullemb
                                : (seq + (size_t)(bb * LL + (tt - 1)) * DD);
  const int col = n0 + l15;

  v8f accS = {};
  v8f accE = {};
  for (int k = 0; k < DD; k += 4) {
    const int ka = k + half * 2;
    v2f a;  a.x  = xrow[ka];                      a.y  = xrow[ka + 1];
    v2f bS; bS.x = Ws[(size_t)ka * HH + col];     bS.y = Ws[(size_t)(ka + 1) * HH + col];
    v2f bE; bE.x = We[(size_t)ka * HH + col];     bE.y = We[(size_t)(ka + 1) * HH + col];
    accS = wmma_f32_k4(a, bS, accS);
    accE = wmma_f32_k4(a, bE, accE);
  }
  const float sb = bs[col], eb = be[col];
#pragma unroll
  for (int r = 0; r < 8; ++r) {
    const int row = m0 + r + half * 8;
    if (row < NROWS) {
      S[(size_t)row * HH + col] = accS[r] + sb;
      E[(size_t)row * HH + col] = accE[r] + eb;
    }
  }
}

// ---------------------------------------------------------------------------
// K2: S1 = S @ W1 ; E1 = E @ W1   (B tile of W1 shared between the two GEMMs)
// ---------------------------------------------------------------------------
__global__ __launch_bounds__(32)
void psf_proj2(const float* __restrict__ S, const float* __restrict__ E,
               const float* __restrict__ W1,
               float* __restrict__ S1, float* __restrict__ E1) {
  const int lane = threadIdx.x & 31;
  const int half = lane >> 4;
  const int l15  = lane & 15;
  const int m0 = blockIdx.x * 16;
  const int n0 = blockIdx.y * 16;

  int g = m0 + l15;
  if (g >= NROWS) g = 0;
  const float* srow = S + (size_t)g * HH;
  const float* erow = E + (size_t)g * HH;
  const int col = n0 + l15;

  v8f accS = {};
  v8f accE = {};
  for (int k = 0; k < HH; k += 4) {
    const int ka = k + half * 2;
    v2f aS; aS.x = srow[ka];                     aS.y = srow[ka + 1];
    v2f aE; aE.x = erow[ka];                     aE.y = erow[ka + 1];
    v2f bv; bv.x = W1[(size_t)ka * HH + col];    bv.y = W1[(size_t)(ka + 1) * HH + col];
    accS = wmma_f32_k4(aS, bv, accS);
    accE = wmma_f32_k4(aE, bv, accE);
  }
#pragma unroll
  for (int r = 0; r < 8; ++r) {
    const int row = m0 + r + half * 8;
    if (row < NROWS) {
      S1[(size_t)row * HH + col] = accS[r];
      E1[(size_t)row * HH + col] = accE[r];
    }
  }
}

// ---------------------------------------------------------------------------
// K3: scores[b,r,j] = mask ? sum_h w2[h]*gelu(e1[b,j,h]-s1[b,r+1,h]+b1[h])+b2
//                          : NEG
// JSPLIT waves cooperate on each output row (b,r), columns j strided by
// JSPLIT -> ~8K waves total so the VALU-bound erf chains are latency-hidden
// by wave interleaving. Lanes split H=256 into 8 contiguous h each;
// s1/b1/w2 folds live in registers, e1 rows are L2-resident float4 loads.
// ---------------------------------------------------------------------------
__global__ __launch_bounds__(256)
void psf_pair(const float* __restrict__ S1, const float* __restrict__ E1,
              const float* __restrict__ b1, const float* __restrict__ w2,
              const float* __restrict__ b2v,
              const unsigned char* __restrict__ seq_mask,
              float* __restrict__ scores) {
  const int lane = threadIdx.x & 31;
  const int wid  = blockIdx.x * (blockDim.x >> 5) + (threadIdx.x >> 5);
  const int c    = wid & (JSPLIT - 1);   // this wave's j-offset
  const int row0 = wid / JSPLIT;         // output row index (b*LL + r)
  const int b = row0 / LL;
  const int r = row0 % LL;
  const int i = r + 1;               // sc[:,1:,:] -> row i of the pairwise grid
  const int h0 = lane * 8;

  float sb[8], w2v[8];
  const float* s1row = S1 + ((size_t)(b * (LL + 1) + i)) * HH + h0;
#pragma unroll
  for (int q = 0; q < 8; ++q) {
    sb[q]  = b1[h0 + q] - s1row[q];  // fold (b1 - s1[i]) once per row
    w2v[q] = w2[h0 + q];
  }
  const bool rowvalid = seq_mask[b * LL + r] != 0;
  const float bias2 = b2v[0];
  float* outrow = scores + (size_t)row0 * (LL + 1);
  const float* e1base = E1 + ((size_t)(b * (LL + 1))) * HH + h0;

  for (int j = c; j <= LL; j += JSPLIT) {
    const float* e1row = e1base + (size_t)j * HH;
    if (j + JSPLIT <= LL)
      __builtin_prefetch(e1row + (size_t)JSPLIT * HH, 0, 0); // global_prefetch
    const float4* e4 = reinterpret_cast<const float4*>(e1row);
    const float4 ea = e4[0];
    const float4 ec = e4[1];
    float xs[8] = {ea.x, ea.y, ea.z, ea.w, ec.x, ec.y, ec.z, ec.w};
    float p = 0.0f;
#pragma unroll
    for (int q = 0; q < 8; ++q)
      p = fmaf(w2v[q], gelu_exact(xs[q] + sb[q]), p);
#pragma unroll
    for (int off = 16; off > 0; off >>= 1)
      p += __shfl_xor(p, off, 32);
    if (lane == 0) {
      outrow[j] = (rowvalid && (r <= j)) ? (p + bias2) : NEGV;
    }
  }
}

// ---------------------------------------------------------------------------
// K4: per-row masked NLL: nll = lse(scores_row) - scores_row[gold]
// One wave per row; deterministic (no atomics).
// ---------------------------------------------------------------------------
__global__ __launch_bounds__(256)
void psf_nll(const float* __restrict__ scores, const int* __restrict__ gold,
             const unsigned char* __restrict__ gmask,
             float* __restrict__ nll, float* __restrict__ cnt) {
  const int lane = threadIdx.x & 31;
  const int wid  = blockIdx.x * (blockDim.x >> 5) + (threadIdx.x >> 5);
  const float* row = scores + (size_t)wid * (LL + 1);

  float mx = -3.4e38f;
  for (int j = lane; j < LL + 1; j += 32) mx = fmaxf(mx, row[j]);
#pragma unroll
  for (int off = 16; off > 0; off >>= 1)
    mx = fmaxf(mx, __shfl_xor(mx, off, 32));

  float se = 0.0f;
  for (int j = lane; j < LL + 1; j += 32) se += expf(row[j] - mx);
#pragma unroll
  for (int off = 16; off > 0; off >>= 1)
    se += __shfl_xor(se, off, 32);

  if (lane == 0) {
    int gi = gold[wid];
    if (gi < 0) gi = 0;
    if (gi > LL) gi = LL;
    const float m = gmask[wid] ? 1.0f : 0.0f;
    const float lse = mx + logf(se);
    nll[wid] = m * (lse - row[gi]);
    cnt[wid] = m;
  }
}

// K5: loss = sum(nll) / sum(mask)
__global__ __launch_bounds__(256)
void psf_loss(const float* __restrict__ nll, const float* __restrict__ cnt,
              float* __restrict__ loss_out) {
  __shared__ float sn[256];
  __shared__ float sm[256];
  const int t = threadIdx.x;
  float a = 0.0f, c = 0.0f;
  for (int idx = t; idx < BD * LL; idx += 256) { a += nll[idx]; c += cnt[idx]; }
  sn[t] = a; sm[t] = c;
  __syncthreads();
  for (int off = 128; off > 0; off >>= 1) {
    if (t < off) { sn[t] += sn[t + off]; sm[t] += sm[t + off]; }
    __syncthreads();
  }
  if (t == 0) loss_out[0] = sn[0] / sm[0];
}

// ---------------------------------------------------------------------------
extern "C" void kernel_launch(void* const* d_in, const int* in_sizes, int n_in,
                              void* d_out, int out_size, void* d_ws, size_t ws_size,
                              hipStream_t stream) {
  const float*         seq      = (const float*)d_in[0];          // [2,512,1024]
  const unsigned char* seq_mask = (const unsigned char*)d_in[1];  // [2,512] bool
  const int*           gold     = (const int*)d_in[2];            // [2,512]
  const unsigned char* gmask    = (const unsigned char*)d_in[3];  // [2,512] bool
  const float*         nullemb  = (const float*)d_in[4];          // [1024]
  const float*         Ws       = (const float*)d_in[5];          // [1024,256]
  const float*         bs       = (const float*)d_in[6];          // [256]
  const float*         We       = (const float*)d_in[7];          // [1024,256]
  const float*         be       = (const float*)d_in[8];          // [256]
  const float*         W1       = (const float*)d_in[9];          // [256,256]
  const float*         b1       = (const float*)d_in[10];         // [256]
  const float*         w2       = (const float*)d_in[11];         // [256]
  const float*         b2v      = (const float*)d_in[12];         // [1]

  float* out = (float*)d_out;               // [2,512,513] scores + [1] loss
  float* ws  = (float*)d_ws;

  const size_t SE = (size_t)NROWS * HH;     // 1026*256 floats
  float* S      = ws;
  float* E      = ws + SE;
  float* S1     = ws + 2 * SE;
  float* E1     = ws + 3 * SE;
  float* nllbuf = ws + 4 * SE;
  float* cntbuf = nllbuf + BD * LL;

  dim3 gproj((NROWS + 15) / 16, HH / 16);   // 65 x 16 tiles, 1 wave each
  psf_proj1<<<gproj, 32, 0, stream>>>(seq, nullemb, Ws, bs, We, be, S, E);
  psf_proj2<<<gproj, 32, 0, stream>>>(S, E, W1, S1, E1);

  // JSPLIT waves per score row: BD*LL*JSPLIT waves / 8 waves per block
  psf_pair<<<(BD * LL * JSPLIT) / 8, 256, 0, stream>>>(S1, E1, b1, w2, b2v,
                                                       seq_mask, out);

  psf_nll<<<(BD * LL) / 8, 256, 0, stream>>>(out, gold, gmask, nllbuf, cntbuf);
  psf_loss<<<1, 256, 0, stream>>>(nllbuf, cntbuf, out + (size_t)BD * LL * (LL + 1));
}